// DistributionWithAttentionPoolingFilter_2516850835894
// MI455X (gfx1250) — compile-verified
//
#include <hip/hip_runtime.h>
#include <stdint.h>

#define Bdim 16
#define Ndim 2048
#define Fdim 128
#define Kbins 21
#define NSPLIT 32
#define NCHUNK (Ndim / NSPLIT)     // 64 n's per workgroup
#define TN 16                      // n-rows per LDS tile
#define NTILES (NCHUNK / TN)       // 4
#define TILE_BYTES (TN * Fdim * 4) // 8192
#define XTOT (Bdim * Fdim * Kbins) // 43008 output elements

// exp(BETA * d^2) = exp2(C2 * d^2), BETA = -50, C2 = BETA * log2(e)
#define C2 (-72.13475204444817f)

typedef float v2f __attribute__((ext_vector_type(2)));
typedef float v8f __attribute__((ext_vector_type(8)));

__device__ __forceinline__ float fast_exp2(float x) {
#if defined(__HIP_DEVICE_COMPILE__) && __has_builtin(__builtin_amdgcn_exp2f)
  return __builtin_amdgcn_exp2f(x); // single v_exp_f32
#else
  return exp2f(x); // HIP math overload (host parse / fallback)
#endif
}

// ---------------------------------------------------------------------------
// Kernel 1: per-(b, n-split) partial weighted Gaussian histograms.
// Data staged global->LDS with CDNA5 async-copy, double buffered.
// Inner loop uses the expanded quadratic so each eval is:
//   v_add + v_fma + v_exp_f32 + v_fma   (constants folded per unrolled k)
// part layout: [s][b*F + f][k]
// ---------------------------------------------------------------------------
__global__ __launch_bounds__(128) void k_partials(
    const float* __restrict__ data, const float* __restrict__ attn,
    float* __restrict__ part) {
  __shared__ __align__(16) float tile[2][TN * Fdim];
  __shared__ float attn_sh[NCHUNK];

  const int tid = threadIdx.x;      // == f
  const int s = blockIdx.x;         // n-split
  const int b = blockIdx.y;
  const int n0 = s * NCHUNK;

  if (tid < NCHUNK) attn_sh[tid] = attn[b * Ndim + n0 + tid];

  const uint64_t gbase =
      (uint64_t)(uintptr_t)(data + ((size_t)b * Ndim + n0) * Fdim);
  const uint32_t l0 = (uint32_t)(uintptr_t)(void*)&tile[0][0];
  const uint32_t l1 = (uint32_t)(uintptr_t)(void*)&tile[1][0];

  auto issue_tile = [&](int t, uint32_t lbase) {
    uint64_t g = gbase + (uint64_t)t * TILE_BYTES + (uint64_t)tid * 16u;
    uint32_t l = lbase + (uint32_t)tid * 16u;
#pragma unroll
    for (int i = 0; i < 4; ++i) {
      // per-lane 16B async copy: 128 lanes x 4 issues = 8 KiB tile
      asm volatile("global_load_async_to_lds_b128 %0, %1, off"
                   :: "v"(l + (uint32_t)(i * 2048)),
                      "v"(g + (uint64_t)(i * 2048))
                   : "memory");
    }
  };

  float acc[Kbins];
#pragma unroll
  for (int k = 0; k < Kbins; ++k) acc[k] = 0.0f;

  issue_tile(0, l0);

  for (int t = 0; t < NTILES; ++t) {
    if (t + 1 < NTILES) {
      issue_tile(t + 1, (t & 1) ? l0 : l1);
      asm volatile("s_wait_asynccnt 4" ::: "memory"); // oldest tile landed
    } else {
      asm volatile("s_wait_asynccnt 0" ::: "memory");
    }
    __syncthreads();

    const float* buf = (t & 1) ? &tile[1][0] : &tile[0][0];
#pragma unroll
    for (int r = 0; r < TN; ++r) {
      const float x = buf[r * Fdim + tid];
      const float w = attn_sh[t * TN + r];
      const float u = C2 * x;  // 2 ops/r amortized over 21 bins
      const float t0 = u * x;  // C2 * x^2
#pragma unroll
      for (int k = 0; k < Kbins; ++k) {
        const float sk = (float)k * 0.05f;         // compile-time constant
        const float ck = -2.0f * C2 * sk;          // folds to literal
        const float ek = C2 * sk * sk;             // folds to literal
        const float arg = fmaf(ck, x, ek) + t0;    // C2*(sk-x)^2
        acc[k] = fmaf(w, fast_exp2(arg), acc[k]);
      }
    }
    __syncthreads(); // all waves done reading before buffer reuse
  }

  float* p = part + (size_t)s * XTOT + ((size_t)(b * Fdim + tid)) * Kbins;
#pragma unroll
  for (int k = 0; k < Kbins; ++k) p[k] = acc[k];
}

// ---------------------------------------------------------------------------
// Kernel 2: reduce partials over the 32 n-splits with WMMA.
// D = ones[16x4] x P[4 x 16] accumulated over 8 chunks -> row 0 holds sums.
// Writes unnormalized out[b*F+f][k].
// ---------------------------------------------------------------------------
__global__ __launch_bounds__(256) void k_reduce(
    const float* __restrict__ part, float* __restrict__ out) {
  const int tid = threadIdx.x;
  const int wave = tid >> 5;
  const int lane = tid & 31;
  const int jt = blockIdx.x * 8 + wave; // 16-element tile of flattened output
  const int j = jt * 16 + (lane & 15);
  const int khalf = lane >> 4;          // which K-pair this half-wave holds

  v2f a = {1.0f, 1.0f}; // all-ones A fragment (16x4)
  v8f acc = {};
#pragma unroll
  for (int c = 0; c < NSPLIT / 4; ++c) {
    const int s0 = c * 4 + 2 * khalf;
    v2f bb;
    bb.x = part[(size_t)s0 * XTOT + j];
    bb.y = part[(size_t)(s0 + 1) * XTOT + j];
    acc = __builtin_amdgcn_wmma_f32_16x16x4_f32(
        false, a, false, bb, (short)0, acc, false, false);
  }
  if (lane < 16) out[j] = acc[0]; // row M=0 of D = column sums
}

// ---------------------------------------------------------------------------
// Kernel 3: normalize each (b,f) row of 21 bins in place. ALFA cancels.
// ---------------------------------------------------------------------------
__global__ __launch_bounds__(256) void k_norm(float* __restrict__ out) {
  const int row = blockIdx.x * blockDim.x + threadIdx.x; // 0..2047
  float* p = out + (size_t)row * Kbins;
  float ssum = 0.0f;
#pragma unroll
  for (int k = 0; k < Kbins; ++k) ssum += p[k];
  const float inv = 1.0f / ssum;
#pragma unroll
  for (int k = 0; k < Kbins; ++k) p[k] *= inv;
}

extern "C" void kernel_launch(void* const* d_in, const int* in_sizes, int n_in,
                              void* d_out, int out_size, void* d_ws,
                              size_t ws_size, hipStream_t stream) {
  (void)in_sizes; (void)n_in; (void)out_size; (void)ws_size;
  const float* data = (const float*)d_in[0];
  const float* attn = (const float*)d_in[1];
  float* out = (float*)d_out;
  float* part = (float*)d_ws; // NSPLIT * 43008 floats = 5.5 MB

  k_partials<<<dim3(NSPLIT, Bdim), 128, 0, stream>>>(data, attn, part);
  k_reduce<<<(XTOT / 16) / 8, 256, 0, stream>>>(part, out); // 336 blocks
  k_norm<<<(Bdim * Fdim) / 256, 256, 0, stream>>>(out);     // 8 blocks
}